// GraphSage_63677185130714
// MI455X (gfx1250) — compile-verified
//
#include <hip/hip_runtime.h>

// ---------------------------------------------------------------------------
// GraphSAGE (max aggregation), 2 layers + log_softmax, for MI455X (gfx1250).
//
// Everything (x, agg1, h, agg2) fits in the 192MB L2, so the edge gather /
// scatter-max phase is L2-bandwidth bound; GEMMs use V_WMMA_F32_16X16X4_F32.
// ---------------------------------------------------------------------------

typedef __attribute__((ext_vector_type(2))) float v2f;
typedef __attribute__((ext_vector_type(8))) float v8f;

#define NEG_INF_BITS 0xFF800000u

// ---------------- fill with -inf bit pattern -------------------------------
__global__ void k_fill_neginf(unsigned int* p, long long n) {
    long long i = (long long)blockIdx.x * blockDim.x + threadIdx.x;
    if (i < n) p[i] = NEG_INF_BITS;
}

// ---------------- scatter max over edges -----------------------------------
// one thread per (edge, feature); F = 1<<shift
__global__ void k_scatter_max(const float* __restrict__ feat,
                              const int* __restrict__ src,
                              const int* __restrict__ dst,
                              float* __restrict__ agg,
                              long long total, int shift, int mask) {
    long long i = (long long)blockIdx.x * blockDim.x + threadIdx.x;
    if (i >= total) return;
    int e = (int)(i >> shift);
    int f = (int)(i & mask);
    int s = src[e];
    int d = dst[e];
    float v = feat[((long long)s << shift) + f];
    float* addr = agg + ((long long)d << shift) + f;
    // monotone int/uint trick for float max (buffer initialized to -inf bits)
    if (v >= 0.0f) atomicMax((int*)addr, __float_as_int(v));
    else           atomicMin((unsigned int*)addr, __float_as_uint(v));
}

// ---------------- pack concatenated weights --------------------------------
// Wcat1[k][n], K=128, N=128 : k<64 -> Wl1[n][k], else Wr1[n][k-64]
__global__ void k_build_w1(const float* __restrict__ Wl,
                           const float* __restrict__ Wr,
                           float* __restrict__ W) {
    int i = blockIdx.x * blockDim.x + threadIdx.x;
    if (i >= 128 * 128) return;
    int k = i >> 7, n = i & 127;
    W[i] = (k < 64) ? Wl[n * 64 + k] : Wr[n * 64 + (k - 64)];
}

// Wcat2[k][n], K=256, N=48 (cols >=40 zero-padded)
__global__ void k_build_w2(const float* __restrict__ Wl,
                           const float* __restrict__ Wr,
                           float* __restrict__ W) {
    int i = blockIdx.x * blockDim.x + threadIdx.x;
    if (i >= 256 * 48) return;
    int k = i / 48, n = i % 48;
    float v = 0.0f;
    if (n < 40) v = (k < 128) ? Wl[n * 128 + k] : Wr[n * 128 + (k - 128)];
    W[i] = v;
}

// ---------------- GEMM layer 1 ---------------------------------------------
// h[16 rows x 128 cols] per block. A = [agg1(-inf->0) | x], K=128.
// 8 waves, each wave one 16x16 tile via v_wmma_f32_16x16x4_f32. ReLU fused.
#define LDS1_STRIDE 132
__global__ __launch_bounds__(256) void k_gemm1(const float* __restrict__ agg,
                                               const float* __restrict__ x,
                                               const float* __restrict__ W,
                                               const float* __restrict__ bias,
                                               float* __restrict__ h,
                                               int nnodes) {
    __shared__ float As[16 * LDS1_STRIDE];
    const int row0 = blockIdx.x * 16;
    const int tid  = threadIdx.x;

    // stage A tile: 16 rows x 128 K (as 32 float4 per row)
    for (int i = tid; i < 16 * 32; i += 256) {
        int r  = i >> 5;
        int k  = (i & 31) << 2;
        int node = row0 + r;
        if (node >= nnodes) node = nnodes - 1;
        float4 v;
        if (k < 64) {
            v = *(const float4*)(agg + (long long)node * 64 + k);
            if (v.x == -__builtin_inff()) v.x = 0.0f;
            if (v.y == -__builtin_inff()) v.y = 0.0f;
            if (v.z == -__builtin_inff()) v.z = 0.0f;
            if (v.w == -__builtin_inff()) v.w = 0.0f;
        } else {
            v = *(const float4*)(x + (long long)node * 64 + (k - 64));
        }
        float* dp = &As[r * LDS1_STRIDE + k];
        dp[0] = v.x; dp[1] = v.y; dp[2] = v.z; dp[3] = v.w;
    }
    __syncthreads();

    const int wave = tid >> 5;
    const int lane = tid & 31;
    const int half = lane >> 4;   // K-pair selector for A/B fragments
    const int m    = lane & 15;   // A row during MAC, output column N index
    const int nb   = wave * 16;   // column base for this wave

    v8f c = {0.f, 0.f, 0.f, 0.f, 0.f, 0.f, 0.f, 0.f};
    for (int k = 0; k < 128; k += 4) {
        int k2 = k + 2 * half;
        v2f a;
        a.x = As[m * LDS1_STRIDE + k2];
        a.y = As[m * LDS1_STRIDE + k2 + 1];
        const float* wp = W + k2 * 128 + nb + m;
        v2f b;
        b.x = wp[0];
        b.y = wp[128];
        c = __builtin_amdgcn_wmma_f32_16x16x4_f32(
                false, a, false, b, (short)0, c, false, false);
    }

    const int col = nb + m;
    const float bv = bias[col];
#pragma unroll
    for (int r = 0; r < 8; ++r) {
        int orow = row0 + r + 8 * half;
        if (orow < nnodes) {
            float v = c[r] + bv;
            h[(long long)orow * 128 + col] = fmaxf(v, 0.0f);  // ReLU
        }
    }
}

// ---------------- GEMM layer 2 ---------------------------------------------
// logits[16 rows x 40 cols] per block. A = [agg2(-inf->0) | h], K=256.
// 3 waves (cols 0..47, >=40 dropped).
#define LDS2_STRIDE 260
__global__ __launch_bounds__(96) void k_gemm2(const float* __restrict__ agg,
                                              const float* __restrict__ hfeat,
                                              const float* __restrict__ W,
                                              const float* __restrict__ bias,
                                              float* __restrict__ logits,
                                              int nnodes) {
    __shared__ float As[16 * LDS2_STRIDE];
    const int row0 = blockIdx.x * 16;
    const int tid  = threadIdx.x;

    for (int i = tid; i < 16 * 64; i += 96) {
        int r  = i >> 6;
        int k  = (i & 63) << 2;
        int node = row0 + r;
        if (node >= nnodes) node = nnodes - 1;
        float4 v;
        if (k < 128) {
            v = *(const float4*)(agg + (long long)node * 128 + k);
            if (v.x == -__builtin_inff()) v.x = 0.0f;
            if (v.y == -__builtin_inff()) v.y = 0.0f;
            if (v.z == -__builtin_inff()) v.z = 0.0f;
            if (v.w == -__builtin_inff()) v.w = 0.0f;
        } else {
            v = *(const float4*)(hfeat + (long long)node * 128 + (k - 128));
        }
        float* dp = &As[r * LDS2_STRIDE + k];
        dp[0] = v.x; dp[1] = v.y; dp[2] = v.z; dp[3] = v.w;
    }
    __syncthreads();

    const int wave = tid >> 5;
    const int lane = tid & 31;
    const int half = lane >> 4;
    const int m    = lane & 15;
    const int nb   = wave * 16;

    v8f c = {0.f, 0.f, 0.f, 0.f, 0.f, 0.f, 0.f, 0.f};
    for (int k = 0; k < 256; k += 4) {
        int k2 = k + 2 * half;
        v2f a;
        a.x = As[m * LDS2_STRIDE + k2];
        a.y = As[m * LDS2_STRIDE + k2 + 1];
        const float* wp = W + k2 * 48 + nb + m;
        v2f b;
        b.x = wp[0];
        b.y = wp[48];
        c = __builtin_amdgcn_wmma_f32_16x16x4_f32(
                false, a, false, b, (short)0, c, false, false);
    }

    const int col = nb + m;
    if (col < 40) {
        const float bv = bias[col];
#pragma unroll
        for (int r = 0; r < 8; ++r) {
            int orow = row0 + r + 8 * half;
            if (orow < nnodes) logits[(long long)orow * 40 + col] = c[r] + bv;
        }
    }
}

// ---------------- log_softmax ----------------------------------------------
__global__ void k_log_softmax(const float* __restrict__ logits,
                              float* __restrict__ out, int n) {
    int i = blockIdx.x * blockDim.x + threadIdx.x;
    if (i >= n) return;
    const float* p = logits + (long long)i * 40;
    float mx = p[0];
#pragma unroll
    for (int j = 1; j < 40; ++j) mx = fmaxf(mx, p[j]);
    float s = 0.0f;
#pragma unroll
    for (int j = 0; j < 40; ++j) s += __expf(p[j] - mx);
    float lse = mx + __logf(s);
    float* o = out + (long long)i * 40;
#pragma unroll
    for (int j = 0; j < 40; ++j) o[j] = p[j] - lse;
}

// ---------------------------------------------------------------------------
extern "C" void kernel_launch(void* const* d_in, const int* in_sizes, int n_in,
                              void* d_out, int out_size, void* d_ws, size_t ws_size,
                              hipStream_t stream) {
    const float* x   = (const float*)d_in[0];
    const int*   ei  = (const int*)d_in[1];
    const float* Wl1 = (const float*)d_in[2];
    const float* bl1 = (const float*)d_in[3];
    const float* Wr1 = (const float*)d_in[4];
    const float* Wl2 = (const float*)d_in[5];
    const float* bl2 = (const float*)d_in[6];
    const float* Wr2 = (const float*)d_in[7];

    const int       N = in_sizes[0] / 64;       // 100000
    const long long E = in_sizes[1] / 2;        // 1600000
    const int* src = ei;
    const int* dst = ei + E;

    // workspace layout (floats)
    float* agg1   = (float*)d_ws;                       // N*64
    float* h      = agg1   + (size_t)N * 64;            // N*128
    float* agg2   = h      + (size_t)N * 128;           // N*128
    float* logits = agg2   + (size_t)N * 128;           // N*40
    float* Wc1    = logits + (size_t)N * 40;            // 128*128
    float* Wc2    = Wc1    + 128 * 128;                 // 256*48

    const int TB = 256;
    const long long n_agg1 = (long long)N * 64;
    const long long n_agg2 = (long long)N * 128;
    const long long e1     = E * 64;
    const long long e2     = E * 128;
    const int nblk16 = (N + 15) / 16;

    // ---- layer 1 ----
    k_fill_neginf<<<(unsigned)((n_agg1 + TB - 1) / TB), TB, 0, stream>>>(
        (unsigned int*)agg1, n_agg1);
    k_build_w1<<<(128 * 128 + TB - 1) / TB, TB, 0, stream>>>(Wl1, Wr1, Wc1);
    k_scatter_max<<<(unsigned)((e1 + TB - 1) / TB), TB, 0, stream>>>(
        x, src, dst, agg1, e1, 6, 63);
    k_gemm1<<<nblk16, 256, 0, stream>>>(agg1, x, Wc1, bl1, h, N);

    // ---- layer 2 ----
    k_fill_neginf<<<(unsigned)((n_agg2 + TB - 1) / TB), TB, 0, stream>>>(
        (unsigned int*)agg2, n_agg2);
    k_build_w2<<<(256 * 48 + TB - 1) / TB, TB, 0, stream>>>(Wl2, Wr2, Wc2);
    k_scatter_max<<<(unsigned)((e2 + TB - 1) / TB), TB, 0, stream>>>(
        h, src, dst, agg2, e2, 7, 127);
    k_gemm2<<<nblk16, 96, 0, stream>>>(agg2, h, Wc2, bl2, logits, N);

    // ---- log_softmax ----
    k_log_softmax<<<(N + TB - 1) / TB, TB, 0, stream>>>(logits, (float*)d_out, N);
}